// VideoAugmentation_2181843386687
// MI455X (gfx1250) — compile-verified
//
#include <hip/hip_runtime.h>
#include <hip/hip_bf16.h>

// Video augmentation pipeline for MI455X (gfx1250, wave32).
// 128 frames of 224x224x3 f32.  Both 77MB ping-pong buffers fit in the 192MB
// L2 together, so read-once streams (videos, noise, last-use of bufA, final
// store) are marked non-temporal to keep the ping-pong set L2-resident.
//
// Pass 1: translate-warp (fused /255)        videos(NT) -> d_out
// Pass 2: zoom-warp                          d_out  -> bufA (ws)
// Pass 3: rotate-warp (angles1)              bufA   -> d_out
// Pass 4: rotate-warp (angles2)              d_out  -> bufA
// Pass 5: y = x + 0.1*noise (float4), chunk partial sums  bufA(NT)+noise(NT) -> d_out
// Pass 6: WMMA (16x16x4 f32, B=ones) reduction -> per-(n,c) means
// Pass 7: contrast remap (float4, NT store, global_prefetch) on d_out

#define HH 224
#define WW 224
#define HW (HH * WW)            // 50176, exact multiple of 256
#define NFRAMES 128
#define CHUNKS 64               // partial-sum chunks per frame
#define CHUNK_PIX (HW / CHUNKS) // 784 pixels = 196 groups of 4

typedef __attribute__((ext_vector_type(2))) float v2f;
typedef __attribute__((ext_vector_type(4))) float v4f;
typedef __attribute__((ext_vector_type(8))) float v8f;

__device__ __forceinline__ int reflect224(int i) {
    // jnp-style reflect, period 2*224 = 448 (coords stay within one period)
    i %= 448;
    if (i < 0) i += 448;
    return (i >= 224) ? (447 - i) : i;
}

// MODE 0: translate (prm = trans[n][2], scaled by H/W), SCALE applies 1/255,
//         NTSRC marks read-once source non-temporal.
// MODE 1: zoom      (prm = zoom[n][2])
// MODE 2: rotate    (prm = angle[n])
template <int MODE, bool SCALE, bool NTSRC>
__global__ void warp_kernel(const float* __restrict__ src,
                            const float* __restrict__ prm,
                            float* __restrict__ dst) {
    int gid = blockIdx.x * 256 + threadIdx.x;       // exact coverage
    int n = gid / HW;                               // block-uniform
    int r = gid - n * HW;
    int yq = r / WW;
    int xq = r - yq * WW;
    float fy = (float)yq;
    float fx = (float)xq;
    const float cy = 111.5f, cx = 111.5f;

    float ys, xs;
    if (MODE == 0) {
        ys = fy - prm[n * 2 + 0] * 224.0f;
        xs = fx - prm[n * 2 + 1] * 224.0f;
    } else if (MODE == 1) {
        ys = cy + (fy - cy) * prm[n * 2 + 0];
        xs = cx + (fx - cx) * prm[n * 2 + 1];
    } else {
        float a = prm[n];
        float ca = cosf(a);
        float sa = sinf(a);
        float dy = fy - cy;
        float dx = fx - cx;
        ys = cy + dy * ca - dx * sa;
        xs = cx + dx * ca + dy * sa;
    }

    float y0f = floorf(ys);
    float x0f = floorf(xs);
    float wy = ys - y0f;
    float wx = xs - x0f;
    int y0 = (int)y0f;
    int x0 = (int)x0f;
    int y0r = reflect224(y0);
    int y1r = reflect224(y0 + 1);
    int x0r = reflect224(x0);
    int x1r = reflect224(x0 + 1);

    const float* f = src + (size_t)n * HW * 3;
    size_t i00 = (size_t)(y0r * WW + x0r) * 3;
    size_t i01 = (size_t)(y0r * WW + x1r) * 3;
    size_t i10 = (size_t)(y1r * WW + x0r) * 3;
    size_t i11 = (size_t)(y1r * WW + x1r) * 3;

    float* o = dst + (size_t)gid * 3;
#pragma unroll
    for (int c = 0; c < 3; ++c) {
        float t00, t01, t10, t11;
        if (NTSRC) {
            t00 = __builtin_nontemporal_load(f + i00 + c);
            t01 = __builtin_nontemporal_load(f + i01 + c);
            t10 = __builtin_nontemporal_load(f + i10 + c);
            t11 = __builtin_nontemporal_load(f + i11 + c);
        } else {
            t00 = f[i00 + c];
            t01 = f[i01 + c];
            t10 = f[i10 + c];
            t11 = f[i11 + c];
        }
        float top = t00 * (1.0f - wx) + t01 * wx;
        float bot = t10 * (1.0f - wx) + t11 * wx;
        float v = top * (1.0f - wy) + bot * wy;
        if (SCALE) v *= (1.0f / 255.0f);
        o[c] = v;
    }
}

// y = x + 0.1*noise with float4 (b128) traffic; per-block partial channel sums.
// grid = NFRAMES*CHUNKS blocks of 256 threads; threads 0..195 each do 4 pixels.
// xin (last use of bufA) and noise (read-once) loaded non-temporally.
__global__ void noise_mean_kernel(const float* __restrict__ xin,
                                  const float* __restrict__ noise,
                                  float* __restrict__ y,
                                  float* __restrict__ partials) {
    int n = blockIdx.x >> 6;
    int chunk = blockIdx.x & 63;
    int tid = threadIdx.x;
    float s0 = 0.0f, s1 = 0.0f, s2 = 0.0f;
    if (tid < CHUNK_PIX / 4) {  // 196 groups of 4 pixels
        size_t fi = ((size_t)n * HW + (size_t)chunk * CHUNK_PIX + (size_t)tid * 4) * 3;
        // 16B-aligned: frame/chunk/group bases are all multiples of 16 bytes
        const v4f* xa = (const v4f*)(xin + fi);
        const v4f* na = (const v4f*)(noise + fi);
        v4f* ya = (v4f*)(y + fi);
        v4f a = __builtin_nontemporal_load(xa + 0) + 0.1f * __builtin_nontemporal_load(na + 0);
        v4f b = __builtin_nontemporal_load(xa + 1) + 0.1f * __builtin_nontemporal_load(na + 1);
        v4f c = __builtin_nontemporal_load(xa + 2) + 0.1f * __builtin_nontemporal_load(na + 2);
        ya[0] = a;  // y is re-read by contrast pass: keep RT (cached)
        ya[1] = b;
        ya[2] = c;
        // channel pattern across 12 floats: a=(0,1,2,0) b=(1,2,0,1) c=(2,0,1,2)
        s0 = a.x + a.w + b.z + c.y;
        s1 = a.y + b.x + b.w + c.z;
        s2 = a.z + b.y + c.x + c.w;
    }
    __shared__ float red[3][256];
    red[0][tid] = s0;
    red[1][tid] = s1;
    red[2][tid] = s2;
    __syncthreads();
    for (int off = 128; off > 0; off >>= 1) {
        if (tid < off) {
            red[0][tid] += red[0][tid + off];
            red[1][tid] += red[1][tid + off];
            red[2][tid] += red[2][tid + off];
        }
        __syncthreads();
    }
    if (tid == 0) {
        partials[(n * 3 + 0) * CHUNKS + chunk] = red[0][0];
        partials[(n * 3 + 1) * CHUNKS + chunk] = red[1][0];
        partials[(n * 3 + 2) * CHUNKS + chunk] = red[2][0];
    }
}

// One wave per frame. Reduce 64 partials/channel with V_WMMA_F32_16X16X4_F32:
// A(16x4) = partials, B(4x16) = all ones  ->  D[m][n] = rowsum(A,m).
// Summing the 8 D regs per lane gives sum(rows 0..7) on lanes 0-15 and
// sum(rows 8..15) on lanes 16-31; two shuffles finish the 64->1 reduction.
__global__ void wmma_mean_kernel(const float* __restrict__ partials,
                                 float* __restrict__ means) {
    int n = blockIdx.x;
    int lane = threadIdx.x;        // single wave32, EXEC all ones at the WMMA
    int m = lane & 15;
    int kb = (lane >> 4) << 1;     // 0 or 2
    v2f b;
    b.x = 1.0f;
    b.y = 1.0f;
#pragma unroll
    for (int c = 0; c < 3; ++c) {
        const float* p = partials + (n * 3 + c) * CHUNKS;
        v2f a;
        a.x = p[m * 4 + kb + 0];
        a.y = p[m * 4 + kb + 1];
        v8f acc = {};
        acc = __builtin_amdgcn_wmma_f32_16x16x4_f32(
            /*neg_a=*/false, a, /*neg_b=*/false, b,
            /*c_mod=*/(short)0, acc, /*reuse_a=*/false, /*reuse_b=*/false);
        float t = acc[0] + acc[1] + acc[2] + acc[3] +
                  acc[4] + acc[5] + acc[6] + acc[7];
        float total = __shfl(t, 0, 32) + __shfl(t, 16, 32);
        if (lane == 0) means[n * 3 + c] = total * (1.0f / (float)HW);
    }
}

// out = (out - mean)*cf + mean, float4 in/out, NT final store.
// 6272 blocks * 256 threads * 4 pixels = all 1,605,632 pixel-groups exactly.
// 3072 floats per block divides the 150528 floats/frame -> n is block-uniform.
__global__ void contrast_kernel(float* __restrict__ y,
                                const float* __restrict__ means,
                                const float* __restrict__ contrast) {
    size_t pix = ((size_t)blockIdx.x * 256 + threadIdx.x) * 4;
    int n = (int)(pix / HW);
    float m0 = means[n * 3 + 0];
    float m1 = means[n * 3 + 1];
    float m2 = means[n * 3 + 2];
    float cf = contrast[n];
    float* p = y + pix * 3;
    __builtin_prefetch((const void*)(p + 12288), 0, 1);  // global_prefetch_b8
    const v4f* pi = (const v4f*)p;
    v4f a = pi[0];
    v4f b = pi[1];
    v4f c = pi[2];
    // channel pattern: a=(0,1,2,0) b=(1,2,0,1) c=(2,0,1,2)
    a.x = (a.x - m0) * cf + m0;  a.y = (a.y - m1) * cf + m1;
    a.z = (a.z - m2) * cf + m2;  a.w = (a.w - m0) * cf + m0;
    b.x = (b.x - m1) * cf + m1;  b.y = (b.y - m2) * cf + m2;
    b.z = (b.z - m0) * cf + m0;  b.w = (b.w - m1) * cf + m1;
    c.x = (c.x - m2) * cf + m2;  c.y = (c.y - m0) * cf + m0;
    c.z = (c.z - m1) * cf + m1;  c.w = (c.w - m2) * cf + m2;
    v4f* po = (v4f*)p;
    __builtin_nontemporal_store(a, po + 0);  // never re-read on device
    __builtin_nontemporal_store(b, po + 1);
    __builtin_nontemporal_store(c, po + 2);
}

extern "C" void kernel_launch(void* const* d_in, const int* in_sizes, int n_in,
                              void* d_out, int out_size, void* d_ws, size_t ws_size,
                              hipStream_t stream) {
    const float* videos   = (const float*)d_in[0];
    const float* trans    = (const float*)d_in[1];
    const float* zoom     = (const float*)d_in[2];
    const float* angles1  = (const float*)d_in[3];
    const float* angles2  = (const float*)d_in[4];
    const float* noise    = (const float*)d_in[5];
    const float* contrast = (const float*)d_in[6];
    float* out = (float*)d_out;

    const size_t NTOT = (size_t)NFRAMES * HW * 3;   // 19,267,584 elements
    float* bufA     = (float*)d_ws;                  // 77 MB ping buffer
    float* partials = bufA + NTOT;                   // 128*3*64 floats
    float* means    = partials + NFRAMES * 3 * CHUNKS;

    const int warpBlocks = (NFRAMES * HW) / 256;     // 25088, exact
    const int vec4Blocks = (int)(NTOT / 12 / 256);   // 6272, exact

    // 1) translate (+ /255), videos read-once -> NT
    warp_kernel<0, true, true><<<warpBlocks, 256, 0, stream>>>(videos, trans, out);
    // 2) zoom                 out -> bufA   (L2-resident ping-pong)
    warp_kernel<1, false, false><<<warpBlocks, 256, 0, stream>>>(out, zoom, bufA);
    // 3) rotate 1             bufA -> out
    warp_kernel<2, false, false><<<warpBlocks, 256, 0, stream>>>(bufA, angles1, out);
    // 4) rotate 2             out -> bufA
    warp_kernel<2, false, false><<<warpBlocks, 256, 0, stream>>>(out, angles2, bufA);
    // 5) +noise (b128, NT in), partial sums   bufA -> out, partials
    noise_mean_kernel<<<NFRAMES * CHUNKS, 256, 0, stream>>>(bufA, noise, out, partials);
    // 6) WMMA final reduction -> means (one wave32 per frame)
    wmma_mean_kernel<<<NFRAMES, 32, 0, stream>>>(partials, means);
    // 7) contrast remap (b128, NT store) in place on out
    contrast_kernel<<<vec4Blocks, 256, 0, stream>>>(out, means, contrast);
}